// ConvPooler_21689584844915
// MI455X (gfx1250) — compile-verified
//
#include <hip/hip_runtime.h>

typedef float v2f __attribute__((ext_vector_type(2)));
typedef float v4f __attribute__((ext_vector_type(4)));
typedef float v8f __attribute__((ext_vector_type(8)));
typedef int   v4i __attribute__((vector_size(16)));   // matches builtin param type

typedef __attribute__((address_space(1))) v4i* glb_v4i_ptr;
typedef __attribute__((address_space(3))) v4i* lds_v4i_ptr;

#define B_        32
#define S_        2048
#define D_        512
#define FULL_     60000

#define TOK_PER_BLOCK 128          // 8 waves x 16 tokens
#define KC            32           // K-chunk per LDS stage
#define NCHUNK        (D_ / KC)    // 16
#define LDS_STRIDE    36           // 32 + 4 pad floats -> conflict-free frag reads

#if __has_builtin(__builtin_amdgcn_global_load_async_to_lds_b128) && \
    __has_builtin(__builtin_amdgcn_s_wait_asynccnt)
#define USE_ASYNC_LDS 1
#else
#define USE_ASYNC_LDS 0
#endif

// ---------------- zero fill: 32*60000 floats = 480000 float4 ----------------
__global__ __launch_bounds__(256)
void ConvPooler_zero_kernel(float* __restrict__ out, int n4) {
    int i = blockIdx.x * blockDim.x + threadIdx.x;
    const int stride = gridDim.x * blockDim.x;
    v4f z = {0.f, 0.f, 0.f, 0.f};
    for (; i < n4; i += stride)
        __builtin_nontemporal_store(z, (v4f*)out + i);
}

// ---------------- GEMV via WMMA + async-LDS staging + scatter ----------------
__global__ __launch_bounds__(256)
void ConvPooler_dot_scatter_kernel(const float* __restrict__ h,
                                   const int*   __restrict__ gene_pos,
                                   const float* __restrict__ w,
                                   const float* __restrict__ bias,
                                   float*       __restrict__ out) {
    __shared__ float lds_w[D_];                                  // 2 KB
    __shared__ float tile[2][TOK_PER_BLOCK * LDS_STRIDE];        // 2 x 18 KB
    __shared__ float lds_vals[TOK_PER_BLOCK];

    const int tid   = threadIdx.x;
    const int lane  = tid & 31;
    const int wave  = tid >> 5;        // 0..7
    const int laneM = lane & 15;       // A-matrix row within the 16-token tile
    const int laneH = lane >> 4;       // 0 -> K=0,1 ; 1 -> K=2,3 (A frag layout)

    const int tokenBase = blockIdx.x * TOK_PER_BLOCK;

    // stage w into LDS once (512 floats, 2 per thread)
    lds_w[tid]       = w[tid];
    lds_w[tid + 256] = w[tid + 256];

    v8f acc = {0.f, 0.f, 0.f, 0.f, 0.f, 0.f, 0.f, 0.f};

#if USE_ASYNC_LDS
    // Async copy engine: global -> LDS directly, tracked by ASYNCcnt.
    // Each chunk = 128 rows x 8 float4 = 1024 b128 transfers, 4 per thread.
    auto issue_chunk = [&](int buf, int k0) {
        #pragma unroll
        for (int i = 0; i < 4; ++i) {
            int idx = tid + i * 256;       // 0..1023
            int row = idx >> 3;
            int c4  = idx & 7;
            const float* g = h + (size_t)(tokenBase + row) * D_ + k0 + c4 * 4;
            float* l = &tile[buf][row * LDS_STRIDE + c4 * 4];
            __builtin_amdgcn_global_load_async_to_lds_b128(
                (glb_v4i_ptr)g, (lds_v4i_ptr)l, 0, 0);
        }
    };

    issue_chunk(0, 0);
    for (int c = 0; c < NCHUNK; ++c) {
        const int cur = c & 1;
        if (c + 1 < NCHUNK) {
            issue_chunk(cur ^ 1, (c + 1) * KC);         // prefetch next chunk
            __builtin_amdgcn_s_wait_asynccnt(4);        // chunk c done (in-order)
        } else {
            __builtin_amdgcn_s_wait_asynccnt(0);
        }
        __syncthreads();

        const float* aRow = &tile[cur][(wave * 16 + laneM) * LDS_STRIDE + 2 * laneH];
        const float* wRow = &lds_w[c * KC + 2 * laneH];
        #pragma unroll
        for (int kk = 0; kk < KC; kk += 4) {
            v2f a = *(const v2f*)(aRow + kk);   // A: 16x4 f32 fragment
            v2f b = *(const v2f*)(wRow + kk);   // B: w[k] broadcast to all 16 cols
            acc = __builtin_amdgcn_wmma_f32_16x16x4_f32(
                      false, a, false, b, (short)0, acc, false, false);
        }
        __syncthreads();   // all waves done with tile[cur] before it is refilled
    }
#else
    // Fallback: staged through VGPRs with non-temporal b128 loads.
    for (int c = 0; c < NCHUNK; ++c) {
        const int cur = c & 1;
        __syncthreads();
        #pragma unroll
        for (int i = 0; i < 4; ++i) {
            int idx = tid + i * 256;
            int row = idx >> 3;
            int c4  = idx & 7;
            v4f v = __builtin_nontemporal_load(
                (const v4f*)(h + (size_t)(tokenBase + row) * D_ + c * KC + c4 * 4));
            *(v4f*)(&tile[cur][row * LDS_STRIDE + c4 * 4]) = v;
        }
        __syncthreads();

        if (c + 1 < NCHUNK) {
            __builtin_prefetch(h + (size_t)(tokenBase + (tid >> 1)) * D_
                                 + (c + 1) * KC + (tid & 1) * 16, 0, 0);
        }

        const float* aRow = &tile[cur][(wave * 16 + laneM) * LDS_STRIDE + 2 * laneH];
        const float* wRow = &lds_w[c * KC + 2 * laneH];
        #pragma unroll
        for (int kk = 0; kk < KC; kk += 4) {
            v2f a = *(const v2f*)(aRow + kk);
            v2f b = *(const v2f*)(wRow + kk);
            acc = __builtin_amdgcn_wmma_f32_16x16x4_f32(
                      false, a, false, b, (short)0, acc, false, false);
        }
    }
#endif

    // D layout: VGPR v -> lanes 0-15 hold M=v, lanes 16-31 hold M=8+v (all N equal)
    __syncthreads();
    if (lane == 0 || lane == 16) {
        int base = wave * 16 + laneH * 8;
        #pragma unroll
        for (int i = 0; i < 8; ++i) lds_vals[base + i] = acc[i];
    }
    __syncthreads();

    if (lane < 16) {
        int tok   = tokenBase + wave * 16 + lane;
        float val = lds_vals[wave * 16 + lane] + bias[0];
        int pos   = gene_pos[tok];
        if (pos >= 0 && pos < FULL_) {
            int bb = tok >> 11;                // tok / S_
            out[(size_t)bb * FULL_ + pos] = val;
        }
    }
}

extern "C" void kernel_launch(void* const* d_in, const int* in_sizes, int n_in,
                              void* d_out, int out_size, void* d_ws, size_t ws_size,
                              hipStream_t stream) {
    const float* h        = (const float*)d_in[0];
    const int*   gene_pos = (const int*)  d_in[1];
    const float* w        = (const float*)d_in[2];
    const float* bias     = (const float*)d_in[3];
    float* out = (float*)d_out;

    const int n4 = (B_ * FULL_) / 4;   // 480000 float4
    ConvPooler_zero_kernel<<<960, 256, 0, stream>>>(out, n4);

    const int nblocks = (B_ * S_) / TOK_PER_BLOCK;   // 512
    ConvPooler_dot_scatter_kernel<<<nblocks, 256, 0, stream>>>(h, gene_pos, w, bias, out);
}